// SimpleNetwork2_9079560864497
// MI455X (gfx1250) — compile-verified
//
#include <hip/hip_runtime.h>
#include <hip/hip_bf16.h>

typedef __attribute__((ext_vector_type(2))) float v2f;
typedef __attribute__((ext_vector_type(8))) float v8f;

#define NEG_SLOPE 0.2f

// ---------- monotonic float<->uint encoding for atomic max ----------
__device__ __forceinline__ unsigned encodeF(float f) {
    unsigned u = __float_as_uint(f);
    return (u >> 31) ? ~u : (u | 0x80000000u);
}
__device__ __forceinline__ float decodeF(unsigned e) {
    unsigned b = (e >> 31) ? (e & 0x7FFFFFFFu) : ~e;
    return __uint_as_float(b);
}
// encodeF(-inf) == ~0xFF800000 == 0x007FFFFF
#define ENC_NEG_INF 0x007FFFFFu

// ---------- WMMA fp32 GEMM: Y[M,Ncol] = X[M,K] @ W[K,Ncol] + bias ----------
// One wave32 per 16x16 output tile; V_WMMA_F32_16X16X4_F32, K stepped by 4.
// Requires M,Ncol multiples of 16 and K multiple of 4. EXEC full per wave.
__global__ void gemm_wmma_f32(const float* __restrict__ X,
                              const float* __restrict__ W,
                              const float* __restrict__ bias,
                              float* __restrict__ Y,
                              int M, int K, int Ncol) {
    const int lane = threadIdx.x & 31;
    const int wave = threadIdx.x >> 5;
    const int wavesPerBlock = blockDim.x >> 5;
    const int tile = blockIdx.x * wavesPerBlock + wave;
    const int tilesN = Ncol >> 4;
    const int tilesM = M >> 4;
    if (tile >= tilesM * tilesN) return;   // whole-wave uniform exit
    const int tm = tile / tilesN;
    const int tn = tile - tm * tilesN;
    const int row0 = tm << 4;
    const int col0 = tn << 4;

    const int half = lane >> 4;   // 0: lanes 0-15, 1: lanes 16-31
    const int l16  = lane & 15;

    v8f acc = {};
    for (int k0 = 0; k0 < K; k0 += 4) {
        const int ka = k0 + half * 2;           // A: vgpr j holds K = 2*half + j
        v2f a, b;
        const float* xrow = X + (size_t)(row0 + l16) * K + ka;
        a.x = xrow[0];
        a.y = xrow[1];
        const float* wrow = W + (size_t)ka * Ncol + col0 + l16;  // B mirrors A's K split
        b.x = wrow[0];
        b.y = wrow[Ncol];
        acc = __builtin_amdgcn_wmma_f32_16x16x4_f32(
            false, a, false, b, (short)0, acc, false, false);
    }
    const float bv = bias ? bias[col0 + l16] : 0.f;
#pragma unroll
    for (int i = 0; i < 8; ++i) {
        const int rm = row0 + i + half * 8;     // C/D: vgpr i -> row i (+8 for hi half)
        Y[(size_t)rm * Ncol + col0 + l16] = acc[i] + bv;
    }
}

// ---------- misc small kernels ----------
__global__ void zero_f(float* p, int n) {
    int i = blockIdx.x * blockDim.x + threadIdx.x;
    if (i < n) p[i] = 0.f;
}

__global__ void init_softmax(unsigned* mx, float* den, int nh) {
    int i = blockIdx.x * blockDim.x + threadIdx.x;
    if (i < nh) { mx[i] = ENC_NEG_INF; den[i] = 0.f; }
}

// pad W[K,Dreal] -> Wd[K,Dpad] with zero columns
__global__ void pad_w(const float* __restrict__ Ws, float* __restrict__ Wd,
                      int K, int Dreal, int Dpad) {
    int i = blockIdx.x * blockDim.x + threadIdx.x;
    if (i >= K * Dpad) return;
    int k = i / Dpad, c = i - k * Dpad;
    Wd[i] = (c < Dreal) ? Ws[k * Dreal + c] : 0.f;
}

// self-loop attr accumulation: deg[dst]+=1, la[dst][:] += ea[e][:]
__global__ void accum_loop(const int* __restrict__ dst,
                           const float* __restrict__ ea,
                           float* __restrict__ la, float* __restrict__ deg, int E) {
    int e = blockIdx.x * blockDim.x + threadIdx.x;
    if (e >= E) return;
    int d = dst[e];
    atomicAdd(&deg[d], 1.f);
#pragma unroll
    for (int j = 0; j < 8; ++j) atomicAdd(&la[d * 8 + j], ea[e * 8 + j]);
}

__global__ void loop_attr_div(float* __restrict__ la, const float* __restrict__ deg, int N) {
    int i = blockIdx.x * blockDim.x + threadIdx.x;
    if (i >= N * 8) return;
    la[i] /= fmaxf(deg[i >> 3], 1.f);
}

// ---------- E1: per-(edge,head) attention logits + segment max ----------
__global__ void edge_logits(const int* __restrict__ srcA, const int* __restrict__ dstA,
                            const float* __restrict__ edge_attr, const float* __restrict__ loop_attr,
                            const float* __restrict__ xl, const float* __restrict__ xr,
                            const float* __restrict__ We, const float* __restrict__ att,
                            float* __restrict__ logits, unsigned* __restrict__ mx,
                            int E, int N, int H, int C, int D, int ldx) {
    int idx = blockIdx.x * blockDim.x + threadIdx.x;
    int total = (E + N) * H;
    if (idx >= total) return;
    int e = idx / H, h = idx - e * H;
    int s, d;
    const float* ea;
    if (e < E) { s = srcA[e]; d = dstA[e]; ea = edge_attr + (size_t)e * 8; }
    else       { s = d = e - E;            ea = loop_attr + (size_t)(e - E) * 8; }
    float ear[8];
#pragma unroll
    for (int j = 0; j < 8; ++j) ear[j] = ea[j];
    const float* xls = xl + (size_t)s * ldx + h * C;
    const float* xrd = xr + (size_t)d * ldx + h * C;
    const float* Wec = We + h * C;
    const float* atc = att + h * C;
    float acc = 0.f;
    for (int c = 0; c < C; ++c) {
        float ep = 0.f;
#pragma unroll
        for (int j = 0; j < 8; ++j) ep = fmaf(ear[j], Wec[j * D + c], ep);
        float v = xls[c] + xrd[c] + ep;
        v = (v > 0.f) ? v : NEG_SLOPE * v;      // leaky_relu
        acc = fmaf(v, atc[c], acc);
    }
    logits[idx] = acc;
    atomicMax(&mx[d * H + h], encodeF(acc));
}

// ---------- E2: exp(logit - max) + segment sum ----------
__global__ void edge_exp(const int* __restrict__ dstA,
                         float* __restrict__ logits, const unsigned* __restrict__ mx,
                         float* __restrict__ den, int E, int N, int H) {
    int idx = blockIdx.x * blockDim.x + threadIdx.x;
    int total = (E + N) * H;
    if (idx >= total) return;
    int e = idx / H, h = idx - e * H;
    int d = (e < E) ? dstA[e] : (e - E);
    float ex = expf(logits[idx] - decodeF(mx[d * H + h]));
    logits[idx] = ex;                            // reuse buffer for ex
    atomicAdd(&den[d * H + h], ex);
}

// ---------- E3: scatter alpha * xl[src] into acc[dst] ----------
__global__ void edge_scatter(const int* __restrict__ srcA, const int* __restrict__ dstA,
                             const float* __restrict__ exb, const float* __restrict__ den,
                             const float* __restrict__ xl, float* __restrict__ acc,
                             int E, int N, int H, int C, int ldx, int ldacc) {
    int idx = blockIdx.x * blockDim.x + threadIdx.x;
    int total = (E + N) * H;
    if (idx >= total) return;
    int e = idx / H, h = idx - e * H;
    int s, d;
    if (e < E) { s = srcA[e]; d = dstA[e]; } else { s = d = e - E; }
    float alpha = exb[idx] / den[d * H + h];
    const float* xls = xl + (size_t)s * ldx + h * C;
    float* o = acc + (size_t)d * ldacc + h * C;
    for (int c = 0; c < C; ++c) atomicAdd(&o[c], alpha * xls[c]);
}

// ---------- E4: bias + optional ReLU (in place) ----------
__global__ void bias_act(float* __restrict__ acc, const float* __restrict__ bias,
                         int N, int D, int ld, int relu) {
    int i = blockIdx.x * blockDim.x + threadIdx.x;
    if (i >= N * D) return;
    int n = i / D, j = i - n * D;
    float v = acc[(size_t)n * ld + j] + bias[j];
    if (relu) v = fmaxf(v, 0.f);
    acc[(size_t)n * ld + j] = v;
}

static inline int cdiv(long a, long b) { return (int)((a + b - 1) / b); }

extern "C" void kernel_launch(void* const* d_in, const int* in_sizes, int n_in,
                              void* d_out, int out_size, void* d_ws, size_t ws_size,
                              hipStream_t stream) {
    const float* x  = (const float*)d_in[0];
    const int*   ei = (const int*)d_in[1];
    const float* ea = (const float*)d_in[2];
    const int Nn = in_sizes[0] / 16;   // 50000
    const int Eg = in_sizes[1] / 2;    // 500000

    const float *Wl[4], *bl[4], *Wr[4], *br[4], *We[4], *att[4], *bias[4];
    for (int i = 0; i < 4; ++i) {
        Wl[i]   = (const float*)d_in[3 + 7 * i];
        bl[i]   = (const float*)d_in[4 + 7 * i];
        Wr[i]   = (const float*)d_in[5 + 7 * i];
        br[i]   = (const float*)d_in[6 + 7 * i];
        We[i]   = (const float*)d_in[7 + 7 * i];
        att[i]  = (const float*)d_in[8 + 7 * i];
        bias[i] = (const float*)d_in[9 + 7 * i];
    }

    // ---- workspace carve ----
    char* w = (char*)d_ws;
    auto allocf = [&](size_t n) { float* p = (float*)w; w += n * sizeof(float); return p; };
    float* xl     = allocf((size_t)Nn * 128);
    float* xr     = allocf((size_t)Nn * 128);
    float* hA     = allocf((size_t)Nn * 128);
    float* hB     = allocf((size_t)Nn * 128);
    float* logits = allocf((size_t)(Eg + Nn) * 4);
    float* den    = allocf((size_t)Nn * 4);
    unsigned* mx  = (unsigned*)allocf((size_t)Nn * 4);
    float* la     = allocf((size_t)Nn * 8);
    float* deg    = allocf((size_t)Nn);
    float* Wl4p   = allocf(128 * 16);
    float* Wr4p   = allocf(128 * 16);
    float* bl4p   = allocf(16);
    float* br4p   = allocf(16);

    const int BT = 256;
    // ---- self-loop attrs: la/deg contiguous -> single zero pass ----
    zero_f<<<cdiv((long)Nn * 9, BT), BT, 0, stream>>>(la, Nn * 9);
    accum_loop<<<cdiv(Eg, BT), BT, 0, stream>>>(ei + Eg, ea, la, deg, Eg);
    loop_attr_div<<<cdiv((long)Nn * 8, BT), BT, 0, stream>>>(la, deg, Nn);
    // ---- pad layer-4 weights/biases to 16 columns ----
    pad_w<<<cdiv(128 * 16, BT), BT, 0, stream>>>(Wl[3], Wl4p, 128, 6, 16);
    pad_w<<<cdiv(128 * 16, BT), BT, 0, stream>>>(Wr[3], Wr4p, 128, 6, 16);
    pad_w<<<1, 32, 0, stream>>>(bl[3], bl4p, 1, 6, 16);
    pad_w<<<1, 32, 0, stream>>>(br[3], br4p, 1, 6, 16);

    const float* hin = x;
    int Kin = 16;
    float* houts[4] = {hA, hB, hA, (float*)d_out};

    for (int L = 0; L < 4; ++L) {
        const int H = (L == 3) ? 1 : 4;
        const int C = (L == 3) ? 6 : 32;
        const int D = H * C;                    // 128 or 6
        const int Npad = (L == 3) ? 16 : 128;   // GEMM Ncol & xl/xr row stride
        const float* WlL = (L == 3) ? Wl4p : Wl[L];
        const float* WrL = (L == 3) ? Wr4p : Wr[L];
        const float* blL = (L == 3) ? bl4p : bl[L];
        const float* brL = (L == 3) ? br4p : br[L];

        // node GEMMs (WMMA): xl = h@Wl+bl ; xr = h@Wr+br
        const int tiles = (Nn >> 4) * (Npad >> 4);
        const int gblocks = cdiv(tiles, 8);     // 8 waves / 256-thread block
        gemm_wmma_f32<<<gblocks, BT, 0, stream>>>(hin, WlL, blL, xl, Nn, Kin, Npad);
        gemm_wmma_f32<<<gblocks, BT, 0, stream>>>(hin, WrL, brL, xr, Nn, Kin, Npad);

        // softmax state + output accumulator init
        init_softmax<<<cdiv((long)Nn * H, BT), BT, 0, stream>>>(mx, den, Nn * H);
        float* acc = houts[L];
        const int ldacc = (L == 3) ? 6 : 128;
        zero_f<<<cdiv((long)Nn * ldacc, BT), BT, 0, stream>>>(acc, Nn * ldacc);

        const int tot = (Eg + Nn) * H;
        edge_logits<<<cdiv(tot, BT), BT, 0, stream>>>(ei, ei + Eg, ea, la, xl, xr,
                                                      We[L], att[L], logits, mx,
                                                      Eg, Nn, H, C, D, Npad);
        edge_exp<<<cdiv(tot, BT), BT, 0, stream>>>(ei + Eg, logits, mx, den, Eg, Nn, H);
        edge_scatter<<<cdiv(tot, BT), BT, 0, stream>>>(ei, ei + Eg, logits, den, xl, acc,
                                                       Eg, Nn, H, C, Npad, ldacc);
        bias_act<<<cdiv((long)Nn * D, BT), BT, 0, stream>>>(acc, bias[L], Nn, D, ldacc,
                                                            (L < 3) ? 1 : 0);
        hin = acc;
        Kin = 128;
    }
    (void)n_in; (void)out_size; (void)ws_size;
}